// Model_21277267984514
// MI455X (gfx1250) — compile-verified
//
#include <hip/hip_runtime.h>
#include <hip/hip_bf16.h>
#include <math.h>

#define NN 50000
#define NE 800000

typedef __attribute__((ext_vector_type(2))) float v2f;
typedef __attribute__((ext_vector_type(8))) float v8f;

// ---------------- degree / normalization ----------------

__global__ void k_init_deg(float* deg, int n) {
    int i = blockIdx.x * blockDim.x + threadIdx.x;
    if (i < n) deg[i] = 1.0f;   // self loop contributes 1
}

__global__ void k_deg_accum(const long long* __restrict__ ei, float* deg, int dstRow) {
    int i = blockIdx.x * blockDim.x + threadIdx.x;
    if (i < NE) {
        int d = (int)ei[(size_t)dstRow * NE + i];
        atomicAdd(&deg[d], 1.0f);
    }
}

__global__ void k_dinv(float* deg, int n) {
    int i = blockIdx.x * blockDim.x + threadIdx.x;
    if (i < n) {
        float d = deg[i];
        deg[i] = d > 0.0f ? rsqrtf(d) : 0.0f;   // in-place: deg -> dinv
    }
}

// ---------------- sparse propagation: out = D^-1/2 (A+I) D^-1/2 h ----------------

// self-loop term initializes the accumulator: out[v] = dinv[v]^2 * h[v]
__global__ void k_selfloop(const float* __restrict__ h, const float* __restrict__ dinv,
                           float* __restrict__ out, int total, int F) {
    int i = blockIdx.x * blockDim.x + threadIdx.x;
    if (i < total) {
        int v = i / F;
        float w = dinv[v];
        out[i] = w * w * h[i];
    }
}

// one wave per edge; lanes stride the feature dimension; scatter with f32 atomics
__global__ void k_edge_prop(const long long* __restrict__ ei, const float* __restrict__ dinv,
                            const float* __restrict__ h, float* __restrict__ out,
                            int F, int srcRow) {
    int gtid = blockIdx.x * blockDim.x + threadIdx.x;
    int edge = gtid >> 5;
    int lane = gtid & 31;
    if (edge >= NE) return;
    int s = (int)ei[(size_t)srcRow * NE + edge];
    int d = (int)ei[(size_t)(1 - srcRow) * NE + edge];
    float w = dinv[s] * dinv[d];
    const float* hs = h + (size_t)s * F;
    float* od = out + (size_t)d * F;
    for (int f = lane; f < F; f += 32)
        atomicAdd(&od[f], w * hs[f]);
}

// ---------------- dense GEMM (WMMA f32 16x16x4) + bias + tanh ----------------
// out[m, colBase + n] = tanh( sum_k H[m,k] * W[k,n] + B[n] ),  n in [0,64), ld(out)=192
// One wave computes a 16-row stripe and all 64 output columns (4 WMMA accumulators).

__global__ void k_gemm_tanh(const float* __restrict__ H, const float* __restrict__ W,
                            const float* __restrict__ B, float* __restrict__ out,
                            int K, int colBase) {
    const int ldo = 192;
    int wave = (blockIdx.x * blockDim.x + threadIdx.x) >> 5;
    int lane = threadIdx.x & 31;
    if (wave >= NN / 16) return;          // wave-uniform guard: EXEC stays all-ones
    int m0   = wave * 16;
    int half = lane >> 4;                 // 0: lanes 0-15, 1: lanes 16-31
    int lm   = lane & 15;

    v8f acc0 = {}, acc1 = {}, acc2 = {}, acc3 = {};
    const float* hrow = H + (size_t)(m0 + lm) * K;

    for (int k0 = 0; k0 < K; k0 += 4) {
        // A fragment: lanes 0-15 -> K = k0+{0,1}; lanes 16-31 -> K = k0+{2,3}
        v2f a;
        a.x = hrow[k0 + 2 * half];
        a.y = hrow[k0 + 2 * half + 1];
        // B fragments (W is K x 64 row-major), same half-lane K split
        const float* wk0 = W + (size_t)(k0 + 2 * half) * 64;
        const float* wk1 = wk0 + 64;
        v2f b0, b1, b2, b3;
        b0.x = wk0[lm];      b0.y = wk1[lm];
        b1.x = wk0[16 + lm]; b1.y = wk1[16 + lm];
        b2.x = wk0[32 + lm]; b2.y = wk1[32 + lm];
        b3.x = wk0[48 + lm]; b3.y = wk1[48 + lm];

        acc0 = __builtin_amdgcn_wmma_f32_16x16x4_f32(false, a, false, b0, (short)0, acc0, false, false);
        acc1 = __builtin_amdgcn_wmma_f32_16x16x4_f32(false, a, false, b1, (short)0, acc1, false, false);
        acc2 = __builtin_amdgcn_wmma_f32_16x16x4_f32(false, a, false, b2, (short)0, acc2, false, false);
        acc3 = __builtin_amdgcn_wmma_f32_16x16x4_f32(false, a, false, b3, (short)0, acc3, false, false);
    }

    // C/D layout: VGPR r -> row m0+r (lanes 0-15) / m0+r+8 (lanes 16-31), col = lm
    int rowBase = m0 + half * 8;
#pragma unroll
    for (int r = 0; r < 8; ++r) {
        float* orow = out + (size_t)(rowBase + r) * ldo + colBase;
        orow[lm]      = tanhf(acc0[r] + B[lm]);
        orow[16 + lm] = tanhf(acc1[r] + B[16 + lm]);
        orow[32 + lm] = tanhf(acc2[r] + B[32 + lm]);
        orow[48 + lm] = tanhf(acc3[r] + B[48 + lm]);
    }
}

// ---------------- host orchestration ----------------

extern "C" void kernel_launch(void* const* d_in, const int* in_sizes, int n_in,
                              void* d_out, int out_size, void* d_ws, size_t ws_size,
                              hipStream_t stream) {
    const float*     x  = (const float*)d_in[0];
    const long long* ei = (const long long*)d_in[1];

    // d_in order: x, edge_index, wout1,bout1,wout2,bout2,wout3,bout3, win1,bin1,win2,bin2,win3,bin3
    const float* w[2][3];
    const float* b[2][3];
    for (int enc = 0; enc < 2; ++enc)
        for (int l = 0; l < 3; ++l) {
            w[enc][l] = (const float*)d_in[2 + enc * 6 + 2 * l];
            b[enc][l] = (const float*)d_in[3 + enc * 6 + 2 * l];
        }

    float* ws   = (float*)d_ws;
    float* dinv = ws;                          // NN
    float* t1   = dinv + NN;                   // NN*256
    float* t2   = t1 + (size_t)NN * 256;       // NN*256
    float* hA   = t2 + (size_t)NN * 256;       // NN*192
    float* hB   = hA + (size_t)NN * 192;       // NN*192

    const int TPB = 256;
    const int gemmBlocks = ((NN / 16) + 7) / 8;   // 8 waves per block, one 16-row tile per wave

    for (int enc = 0; enc < 2; ++enc) {
        int dstRow = (enc == 0) ? 1 : 0;   // "in" encoder uses reversed edge_index
        int srcRow = 1 - dstRow;
        float* zfinal = (float*)d_out + (size_t)enc * NN * 192;

        // normalization
        k_init_deg<<<(NN + TPB - 1) / TPB, TPB, 0, stream>>>(dinv, NN);
        k_deg_accum<<<(NE + TPB - 1) / TPB, TPB, 0, stream>>>(ei, dinv, dstRow);
        k_dinv<<<(NN + TPB - 1) / TPB, TPB, 0, stream>>>(dinv, NN);

        const float* h = x;
        int din = 256;
        for (int layer = 0; layer < 3; ++layer) {
            float* outbuf = (layer == 2) ? zfinal : ((layer == 0) ? hA : hB);

            // t1 = prop(h); t2 = prop(t1)
            int total = NN * din;
            k_selfloop<<<(total + TPB - 1) / TPB, TPB, 0, stream>>>(h, dinv, t1, total, din);
            k_edge_prop<<<(NE * 32) / TPB, TPB, 0, stream>>>(ei, dinv, h, t1, din, srcRow);
            k_selfloop<<<(total + TPB - 1) / TPB, TPB, 0, stream>>>(t1, dinv, t2, total, din);
            k_edge_prop<<<(NE * 32) / TPB, TPB, 0, stream>>>(ei, dinv, t1, t2, din, srcRow);

            // three 64-wide linear maps into the 192-wide concat buffer, fused bias+tanh
            for (int p = 0; p < 3; ++p) {
                const float* src = (p == 0) ? h : ((p == 1) ? t1 : t2);
                k_gemm_tanh<<<gemmBlocks, TPB, 0, stream>>>(
                    src, w[enc][layer] + (size_t)p * din * 64, b[enc][layer] + p * 64,
                    outbuf, din, p * 64);
            }

            h = outbuf;
            din = 192;
        }
    }
}